// GetDensity_13718125543713
// MI455X (gfx1250) — compile-verified
//
#include <hip/hip_runtime.h>
#include <math.h>

// ---- fixed problem shape from the reference -------------------------------
#define NB     16                 // nbatch
#define NA     512                // numatom
#define NEIGH  64
#define PPM    (NA * NEIGH)       // 32768 pairs per batch
#define NT     4                  // ntype
#define NW     8                  // nwave
#define NCOMP  13                 // 1 + 3 + 9 angular components
#define NLVL   3                  // NIPSIN
#define CUTF   5.0f

#define SPLIT  8                  // pair-chunks per batch (occupancy)
#define CHUNK  (PPM / SPLIT)      // 4096 pairs per workgroup
#define SPAD   105                // per-atom LDS stride in dwords; odd => coprime
                                  // with the 64 LDS banks => conflict-free scatter
#define SUMW_ELEMS ((size_t)NB * NA * NCOMP * NW)   // global sum_w floats

typedef __attribute__((ext_vector_type(2))) float v2f;
typedef __attribute__((ext_vector_type(8))) float v8f;

__device__ __forceinline__ int level_of(int c) {
    // index_para: comp0 -> 0, comps1..3 -> 1, comps4..12 -> 2, >=13 pad -> -1
    return (c == 0) ? 0 : (c < 4) ? 1 : (c < NCOMP) ? 2 : -1;
}

// Per-pair physics: returns 13 angular components and 8 weighted radials.
__device__ __forceinline__ void pair_terms(
    const float* __restrict__ s_cart, const int* __restrict__ s_spec,
    const float* __restrict__ s_rs, const float* __restrict__ s_inta,
    const float* __restrict__ s_prm,
    int i0, int j0, float sh0, float sh1, float sh2,
    float ang[NCOMP], float w[NW])
{
    const float pi_over_cut = 3.14159265358979323846f / CUTF;
    const float valid = (sh0 > -1.0e10f && sh1 > -1.0e10f && sh2 > -1.0e10f) ? 1.0f : 0.0f;
    const float dx = s_cart[i0 * 3 + 0] - s_cart[j0 * 3 + 0] + sh0;
    const float dy = s_cart[i0 * 3 + 1] - s_cart[j0 * 3 + 1] + sh1;
    const float dz = s_cart[i0 * 3 + 2] - s_cart[j0 * 3 + 2] + sh2;
    const float d  = sqrtf(dx * dx + dy * dy + dz * dz);
    const float cf   = 0.5f * __cosf(d * pi_over_cut) + 0.5f;
    const float fcut = cf * cf * valid;
    const int sp = s_spec[j0];
    #pragma unroll
    for (int n = 0; n < NW; ++n) {
        const float r = d - s_rs[sp * NW + n];
        w[n] = __expf(-s_inta[sp * NW + n] * r * r) * s_prm[sp * NW + n];
    }
    ang[0] = fcut;
    ang[1] = fcut * dx;  ang[2] = fcut * dy;  ang[3] = fcut * dz;
    ang[4]  = ang[1] * dx; ang[5]  = ang[1] * dy; ang[6]  = ang[1] * dz;
    ang[7]  = ang[2] * dx; ang[8]  = ang[2] * dy; ang[9]  = ang[2] * dz;
    ang[10] = ang[3] * dx; ang[11] = ang[3] * dy; ang[12] = ang[3] * dz;
}

// ===========================================================================
// Path A (3 kernels, needs ws_size >= NB*NA*104*4 bytes): high occupancy.
// ===========================================================================

__global__ __launch_bounds__(256, 4)
void zero_ws_kernel(float* __restrict__ ws) {
    const size_t i = (size_t)blockIdx.x * blockDim.x + threadIdx.x;
    if (i < SUMW_ELEMS) ws[i] = 0.0f;
}

// 128 workgroups: block = (batch, chunk). Private 215KB LDS partial
// accumulator (CDNA5 320KB/WGP), ds_add_f32 scatter, then global-atomic merge.
__global__ __launch_bounds__(512, 1)
void accumulate_kernel(const float* __restrict__ cart,
                       const int*   __restrict__ species,
                       const int*   __restrict__ atom_index,
                       const float* __restrict__ shifts,
                       const float* __restrict__ rs,
                       const float* __restrict__ inta,
                       const float* __restrict__ prm,
                       float*       __restrict__ ws)      // (NB*NA, 104)
{
    __shared__ float s_sum[NA * SPAD];     // 53760 f32 = 215040 B, odd stride
    __shared__ float s_cart[NA * 3];
    __shared__ int   s_spec[NA];
    __shared__ float s_rs[NT * NW], s_inta[NT * NW], s_prm[NT * NW];

    const int b     = blockIdx.x / SPLIT;
    const int chunk = blockIdx.x % SPLIT;
    const int t     = threadIdx.x;
    const int nthr  = blockDim.x;

    for (int i = t; i < NA * SPAD; i += nthr) s_sum[i] = 0.0f;
    for (int i = t; i < NA * 3;    i += nthr) s_cart[i] = cart[(size_t)b * NA * 3 + i];
    for (int i = t; i < NA;        i += nthr) s_spec[i] = species[b * NA + i];
    if (t < NT * NW) { s_rs[t] = rs[t]; s_inta[t] = inta[t]; s_prm[t] = prm[t]; }
    __syncthreads();

    for (int p = chunk * CHUNK + t; p < (chunk + 1) * CHUNK; p += nthr) {
        const int i0 = atom_index[(size_t)(0 * NB + b) * PPM + p];
        const int j0 = atom_index[(size_t)(1 * NB + b) * PPM + p];
        const size_t shb = ((size_t)b * PPM + p) * 3;
        float ang[NCOMP], w[NW];
        pair_terms(s_cart, s_spec, s_rs, s_inta, s_prm,
                   i0, j0, shifts[shb], shifts[shb + 1], shifts[shb + 2], ang, w);
        float* base = &s_sum[i0 * SPAD];
        #pragma unroll
        for (int c = 0; c < NCOMP; ++c)
            #pragma unroll
            for (int n = 0; n < NW; ++n)
                atomicAdd(&base[c * NW + n], ang[c] * w[n]);   // ds_add_f32
    }
    __syncthreads();

    // merge partial accumulator into global sum_w
    for (int e = t; e < NA * NCOMP * NW; e += nthr) {
        const int a = e / (NCOMP * NW);
        const int r = e - a * (NCOMP * NW);
        atomicAdd(&ws[((size_t)b * NA + a) * (NCOMP * NW) + r], s_sum[a * SPAD + r]);
    }
}

// 512 waves: one 16-atom group per wave. density = (sum_w)^2 @ level-indicator
// as 4 chained V_WMMA_F32_16X16X4_F32 per (group, n).
__global__ __launch_bounds__(128, 2)
void reduce_wmma_kernel(const float* __restrict__ ws, float* __restrict__ out)
{
    const int t    = threadIdx.x;
    const int lane = t & 31;
    const int row  = lane & 15;
    const int half = lane >> 4;
    const int group = blockIdx.x * (blockDim.x >> 5) + (t >> 5);  // 0..511

    v2f bfrag[4];
    #pragma unroll
    for (int cb4 = 0; cb4 < 4; ++cb4) {
        const int cA = cb4 * 4 + (half ? 2 : 0);
        const int cB = cb4 * 4 + (half ? 3 : 1);
        bfrag[cb4].x = (level_of(cA) == row) ? 1.0f : 0.0f;
        bfrag[cb4].y = (level_of(cB) == row) ? 1.0f : 0.0f;
    }

    const size_t aRow = (size_t)group * 16 + row;   // this lane's A-matrix row
    #pragma unroll
    for (int n = 0; n < NW; ++n) {
        v8f acc = {0.f, 0.f, 0.f, 0.f, 0.f, 0.f, 0.f, 0.f};
        #pragma unroll
        for (int cb4 = 0; cb4 < 4; ++cb4) {
            const int cA = cb4 * 4 + (half ? 2 : 0);
            const int cB = cb4 * 4 + (half ? 3 : 1);
            // branchless K-padding: clamped address + mask multiply
            const int ccA = (cA < NCOMP) ? cA : 0;
            const int ccB = (cB < NCOMP) ? cB : 0;
            const float mA = (cA < NCOMP) ? 1.0f : 0.0f;
            const float mB = (cB < NCOMP) ? 1.0f : 0.0f;
            const float vA = ws[aRow * (NCOMP * NW) + ccA * NW + n] * mA;
            const float vB = ws[aRow * (NCOMP * NW) + ccB * NW + n] * mB;
            v2f afrag; afrag.x = vA * vA; afrag.y = vB * vB;
            acc = __builtin_amdgcn_wmma_f32_16x16x4_f32(
                false, afrag, false, bfrag[cb4], (short)0, acc, false, false);
        }
        if (row < NLVL) {
            #pragma unroll
            for (int r = 0; r < 8; ++r) {
                const size_t atom = (size_t)group * 16 + r + (half ? 8 : 0);
                out[atom * (NLVL * NW) + row * NW + n] = acc[r];
            }
        }
    }
}

// ===========================================================================
// Path B fallback (single kernel, no workspace needed): round-1 design with
// odd-stride LDS padding and branchless WMMA feed.
// ===========================================================================
__global__ __launch_bounds__(512, 1)
void fused_kernel(const float* __restrict__ cart,
                  const int*   __restrict__ species,
                  const int*   __restrict__ atom_index,
                  const float* __restrict__ shifts,
                  const float* __restrict__ rs,
                  const float* __restrict__ inta,
                  const float* __restrict__ prm,
                  float*       __restrict__ out)
{
    __shared__ float s_sum[NA * SPAD];
    __shared__ float s_cart[NA * 3];
    __shared__ int   s_spec[NA];
    __shared__ float s_rs[NT * NW], s_inta[NT * NW], s_prm[NT * NW];

    const int b    = blockIdx.x;
    const int t    = threadIdx.x;
    const int nthr = blockDim.x;

    for (int i = t; i < NA * SPAD; i += nthr) s_sum[i] = 0.0f;
    for (int i = t; i < NA * 3;    i += nthr) s_cart[i] = cart[(size_t)b * NA * 3 + i];
    for (int i = t; i < NA;        i += nthr) s_spec[i] = species[b * NA + i];
    if (t < NT * NW) { s_rs[t] = rs[t]; s_inta[t] = inta[t]; s_prm[t] = prm[t]; }
    __syncthreads();

    for (int p = t; p < PPM; p += nthr) {
        const int i0 = atom_index[(size_t)(0 * NB + b) * PPM + p];
        const int j0 = atom_index[(size_t)(1 * NB + b) * PPM + p];
        const size_t shb = ((size_t)b * PPM + p) * 3;
        float ang[NCOMP], w[NW];
        pair_terms(s_cart, s_spec, s_rs, s_inta, s_prm,
                   i0, j0, shifts[shb], shifts[shb + 1], shifts[shb + 2], ang, w);
        float* base = &s_sum[i0 * SPAD];
        #pragma unroll
        for (int c = 0; c < NCOMP; ++c)
            #pragma unroll
            for (int n = 0; n < NW; ++n)
                atomicAdd(&base[c * NW + n], ang[c] * w[n]);
    }
    __syncthreads();

    const int wave = t >> 5;
    const int lane = t & 31;
    const int row  = lane & 15;
    const int half = lane >> 4;

    v2f bfrag[4];
    #pragma unroll
    for (int cb4 = 0; cb4 < 4; ++cb4) {
        const int cA = cb4 * 4 + (half ? 2 : 0);
        const int cB = cb4 * 4 + (half ? 3 : 1);
        bfrag[cb4].x = (level_of(cA) == row) ? 1.0f : 0.0f;
        bfrag[cb4].y = (level_of(cB) == row) ? 1.0f : 0.0f;
    }

    for (int g = wave * 2; g < wave * 2 + 2; ++g) {
        const int    aL = g * 16 + row;
        const size_t a0 = (size_t)b * NA + g * 16;
        #pragma unroll
        for (int n = 0; n < NW; ++n) {
            v8f acc = {0.f, 0.f, 0.f, 0.f, 0.f, 0.f, 0.f, 0.f};
            #pragma unroll
            for (int cb4 = 0; cb4 < 4; ++cb4) {
                const int cA = cb4 * 4 + (half ? 2 : 0);
                const int cB = cb4 * 4 + (half ? 3 : 1);
                const int ccA = (cA < NCOMP) ? cA : 0;
                const int ccB = (cB < NCOMP) ? cB : 0;
                const float mA = (cA < NCOMP) ? 1.0f : 0.0f;
                const float mB = (cB < NCOMP) ? 1.0f : 0.0f;
                const float vA = s_sum[aL * SPAD + ccA * NW + n] * mA;
                const float vB = s_sum[aL * SPAD + ccB * NW + n] * mB;
                v2f afrag; afrag.x = vA * vA; afrag.y = vB * vB;
                acc = __builtin_amdgcn_wmma_f32_16x16x4_f32(
                    false, afrag, false, bfrag[cb4], (short)0, acc, false, false);
            }
            if (row < NLVL) {
                #pragma unroll
                for (int r = 0; r < 8; ++r) {
                    const size_t atom = a0 + r + (half ? 8 : 0);
                    out[atom * (NLVL * NW) + row * NW + n] = acc[r];
                }
            }
        }
    }
}

extern "C" void kernel_launch(void* const* d_in, const int* in_sizes, int n_in,
                              void* d_out, int out_size, void* d_ws, size_t ws_size,
                              hipStream_t stream) {
    (void)in_sizes; (void)n_in; (void)out_size;
    const float* cart       = (const float*)d_in[0];
    // d_in[1] = numatoms (all == NA) -- unused
    const int*   species    = (const int*)  d_in[2];
    const int*   atom_index = (const int*)  d_in[3];
    const float* shifts     = (const float*)d_in[4];
    const float* rs         = (const float*)d_in[5];
    const float* inta       = (const float*)d_in[6];
    const float* prm        = (const float*)d_in[7];
    float*       out        = (float*)d_out;

    const size_t need = SUMW_ELEMS * sizeof(float);   // 3.25 MiB
    if (d_ws != nullptr && ws_size >= need) {
        float* ws = (float*)d_ws;
        const int zthreads = 256;
        const int zblocks  = (int)((SUMW_ELEMS + zthreads - 1) / zthreads);
        zero_ws_kernel<<<zblocks, zthreads, 0, stream>>>(ws);
        accumulate_kernel<<<NB * SPLIT, 512, 0, stream>>>(
            cart, species, atom_index, shifts, rs, inta, prm, ws);
        reduce_wmma_kernel<<<128, 128, 0, stream>>>(ws, out);
    } else {
        fused_kernel<<<NB, 512, 0, stream>>>(
            cart, species, atom_index, shifts, rs, inta, prm, out);
    }
}